// TanhAttention_22539988369470
// MI455X (gfx1250) — compile-verified
//
#include <hip/hip_runtime.h>

#define D_DIM  512
#define B_DIM  16
#define TV_DIM 256
#define TS_DIM 32
#define VPB    8    // v-rows per block in tanh kernel

#define KC     32   // k-chunk depth staged in LDS
#define LDS_S  36   // LDS row stride (KC + 4 pad): b64-aligned, bank-conflict-free frags

typedef float v2f __attribute__((ext_vector_type(2)));
typedef float v8f __attribute__((ext_vector_type(8)));

__device__ __forceinline__ float fast_tanh(float x) {
#if __has_builtin(__builtin_amdgcn_tanhf)
  return __builtin_amdgcn_tanhf(x);
#else
  float y;
  asm("v_tanh_f32 %0, %1" : "=v"(y) : "v"(x));
  return y;
#endif
}

// out[m,n] = (sum_k A[m,k] * W[n,k] + bias[n]) * rmask[m]
// Block (256 thr = 8 waves) computes a 128x64 tile; wave w owns rows [16w,16w+16),
// all waves share the 64 W-columns staged in LDS. K processed in KC-deep chunks,
// double-buffered: chunk c+1 is loaded global->regs during compute of chunk c.
__global__ __launch_bounds__(256) void gemm_xwt_wmma(
    const float* __restrict__ A, const float* __restrict__ W,
    const float* __restrict__ bias, const float* __restrict__ rmask,
    float* __restrict__ out, int M, int N, int K) {
  __shared__ float As[2][128 * LDS_S];   // 2 x 18432 B
  __shared__ float Ws[2][ 64 * LDS_S];   // 2 x  9216 B

  const int tid  = threadIdx.x;
  const int lane = tid & 31;
  const int wib  = tid >> 5;
  const int ntn  = N >> 6;
  const int mblk = blockIdx.x / ntn;
  const int ng   = blockIdx.x % ntn;
  const int mBase = mblk * 128;
  const int n0    = ng * 64;
  const int half  = lane >> 4;           // 0: K pair {0,1}, 1: K pair {2,3}
  const int lm    = lane & 15;
  const int nchunks = K / KC;

  // per-thread global->LDS staging coordinates (float4 granularity)
  int arR[4], arC[4], wrR[2], wrC[2];
#pragma unroll
  for (int i = 0; i < 4; ++i) { int f = tid + i * 256; arR[i] = f >> 3; arC[i] = (f & 7) * 4; }
#pragma unroll
  for (int i = 0; i < 2; ++i) { int f = tid + i * 256; wrR[i] = f >> 3; wrC[i] = (f & 7) * 4; }

  float4 ar[4], wr[2];
  // prologue: fetch chunk 0
#pragma unroll
  for (int i = 0; i < 4; ++i)
    ar[i] = *(const float4*)(A + (size_t)(mBase + arR[i]) * K + arC[i]);
#pragma unroll
  for (int i = 0; i < 2; ++i)
    wr[i] = *(const float4*)(W + (size_t)(n0 + wrR[i]) * K + wrC[i]);
#pragma unroll
  for (int i = 0; i < 4; ++i)
    *(float4*)&As[0][arR[i] * LDS_S + arC[i]] = ar[i];
#pragma unroll
  for (int i = 0; i < 2; ++i)
    *(float4*)&Ws[0][wrR[i] * LDS_S + wrC[i]] = wr[i];
  __syncthreads();

  v8f c0 = {0.f,0.f,0.f,0.f,0.f,0.f,0.f,0.f};
  v8f c1 = {0.f,0.f,0.f,0.f,0.f,0.f,0.f,0.f};
  v8f c2 = {0.f,0.f,0.f,0.f,0.f,0.f,0.f,0.f};
  v8f c3 = {0.f,0.f,0.f,0.f,0.f,0.f,0.f,0.f};

  const int arow = 16 * wib + lm;
  for (int c = 0; c < nchunks; ++c) {
    const int buf = c & 1;
    const int nbuf = buf ^ 1;
    const bool more = (c + 1) < nchunks;
    if (more) {                                      // issue next chunk's global loads
      const int k0n = (c + 1) * KC;
#pragma unroll
      for (int i = 0; i < 4; ++i)
        ar[i] = *(const float4*)(A + (size_t)(mBase + arR[i]) * K + k0n + arC[i]);
#pragma unroll
      for (int i = 0; i < 2; ++i)
        wr[i] = *(const float4*)(W + (size_t)(n0 + wrR[i]) * K + k0n + wrC[i]);
    }
    // compute chunk c out of LDS
    const float* Ab = &As[buf][arow * LDS_S + 2 * half];
    const float* Wb = &Ws[buf][lm   * LDS_S + 2 * half];
#pragma unroll
    for (int kk = 0; kk < KC; kk += 4) {
      v2f a  = *(const v2f*)(Ab + kk);
      v2f b0 = *(const v2f*)(Wb + kk);
      v2f b1 = *(const v2f*)(Wb + 16 * LDS_S + kk);
      v2f b2 = *(const v2f*)(Wb + 32 * LDS_S + kk);
      v2f b3 = *(const v2f*)(Wb + 48 * LDS_S + kk);
      c0 = __builtin_amdgcn_wmma_f32_16x16x4_f32(false, a, false, b0, (short)0, c0, false, false);
      c1 = __builtin_amdgcn_wmma_f32_16x16x4_f32(false, a, false, b1, (short)0, c1, false, false);
      c2 = __builtin_amdgcn_wmma_f32_16x16x4_f32(false, a, false, b2, (short)0, c2, false, false);
      c3 = __builtin_amdgcn_wmma_f32_16x16x4_f32(false, a, false, b3, (short)0, c3, false, false);
    }
    if (more) {                                      // stash next chunk into other buffer
#pragma unroll
      for (int i = 0; i < 4; ++i)
        *(float4*)&As[nbuf][arR[i] * LDS_S + arC[i]] = ar[i];
#pragma unroll
      for (int i = 0; i < 2; ++i)
        *(float4*)&Ws[nbuf][wrR[i] * LDS_S + wrC[i]] = wr[i];
    }
    __syncthreads();
  }

  // C/D layout: VGPR r holds M = m0 + r + 8*half, N = n0 + lm (+16j)
  const int m0  = mBase + 16 * wib;
  const int col = n0 + lm;
  const float bb0 = bias ? bias[col +  0] : 0.f;
  const float bb1 = bias ? bias[col + 16] : 0.f;
  const float bb2 = bias ? bias[col + 32] : 0.f;
  const float bb3 = bias ? bias[col + 48] : 0.f;
#pragma unroll
  for (int r = 0; r < 8; ++r) {
    const int row  = m0 + r + 8 * half;
    const float rm = rmask[row];
    float* orow = out + (size_t)row * N;
    orow[col +  0] = (c0[r] + bb0) * rm;
    orow[col + 16] = (c1[r] + bb1) * rm;
    orow[col + 32] = (c2[r] + bb2) * rm;
    orow[col + 48] = (c3[r] + bb3) * rm;
  }
}

// out[b,v,s] = pm * sum_d tanh((tmp1[b,v,d] + tmp2[b,s,d]) * pm) * wt[d],  pm = vm[b,v]*sm[b,s]
// Block = (b, 8 v-rows), one wave per v-row, lane owns s. tmp2[b] staged transposed in LDS
// with stride-33 padding (conflict-free), in two 256-deep d-halves (52 KB LDS total).
__global__ __launch_bounds__(256) void tanh_dot_kernel(
    const float* __restrict__ t1g,   // [B*TV, D]
    const float* __restrict__ t2g,   // [B*TS, D]
    const float* __restrict__ wtg,   // [D]
    const float* __restrict__ vmask, // [B*TV]
    const float* __restrict__ smask, // [B*TS]
    float* __restrict__ out) {       // [B*TV*TS]
  __shared__ float t2s[256 * 33];                    // d-half x 32 s, padded
  __shared__ __align__(16) float wts[D_DIM];
  __shared__ __align__(16) float t1s[VPB * D_DIM];

  const int tid  = threadIdx.x;
  const int lane = tid & 31;                         // = s
  const int w    = tid >> 5;                         // wave -> v-row
  const int b    = blockIdx.x / (TV_DIM / VPB);
  const int v    = (blockIdx.x % (TV_DIM / VPB)) * VPB + w;
  const int rowv = b * TV_DIM + v;

  for (int i = tid; i < D_DIM; i += 256) wts[i] = wtg[i];
#pragma unroll
  for (int j = 0; j < D_DIM / 32; ++j)
    t1s[w * D_DIM + lane + 32 * j] = t1g[(size_t)rowv * D_DIM + lane + 32 * j];

  const float pm = vmask[rowv] * smask[b * TS_DIM + lane];
  float acc = 0.f;

  for (int h = 0; h < 2; ++h) {
    if (h) __syncthreads();                          // drain compute before overwrite
    for (int i = tid; i < 256 * TS_DIM; i += 256) {  // coalesced: 256-float runs per s
      const int s  = i >> 8;
      const int dl = i & 255;
      t2s[dl * 33 + s] = t2g[(size_t)(b * TS_DIM + s) * D_DIM + h * 256 + dl];
    }
    __syncthreads();

    const float* t1h = t1s + w * D_DIM + h * 256;
    const float* wth = wts + h * 256;
#pragma unroll 4
    for (int d = 0; d < 256; d += 4) {
      const float4 x1 = *(const float4*)(t1h + d);   // ds_load_b128 broadcast
      const float4 xw = *(const float4*)(wth + d);
      const float a0 = t2s[(d + 0) * 33 + lane];
      const float a1 = t2s[(d + 1) * 33 + lane];
      const float a2 = t2s[(d + 2) * 33 + lane];
      const float a3 = t2s[(d + 3) * 33 + lane];
      acc = fmaf(fast_tanh((x1.x + a0) * pm), xw.x, acc);
      acc = fmaf(fast_tanh((x1.y + a1) * pm), xw.y, acc);
      acc = fmaf(fast_tanh((x1.z + a2) * pm), xw.z, acc);
      acc = fmaf(fast_tanh((x1.w + a3) * pm), xw.w, acc);
    }
  }

  out[(size_t)rowv * TS_DIM + lane] = acc * pm;      // coalesced 32-float row store
}

extern "C" void kernel_launch(void* const* d_in, const int* in_sizes, int n_in,
                              void* d_out, int out_size, void* d_ws, size_t ws_size,
                              hipStream_t stream) {
  const float* video = (const float*)d_in[0];
  const float* vmask = (const float*)d_in[1];
  const float* sent  = (const float*)d_in[2];
  const float* smask = (const float*)d_in[3];
  const float* w1    = (const float*)d_in[4];
  const float* b1    = (const float*)d_in[5];
  const float* w2    = (const float*)d_in[6];
  const float* wt    = (const float*)d_in[7];
  float* out  = (float*)d_out;

  float* tmp1 = (float*)d_ws;                                  // [4096, 512] = 8 MB
  float* tmp2 = tmp1 + (size_t)B_DIM * TV_DIM * D_DIM;         // [512, 512]  = 1 MB

  // tmp1: (4096/128) m-blocks x (512/64) n-groups = 256 blocks of 8 waves
  gemm_xwt_wmma<<<256, 256, 0, stream>>>(video, w1, b1, vmask, tmp1,
                                         B_DIM * TV_DIM, D_DIM, D_DIM);
  // tmp2: (512/128) x 8 = 32 blocks
  gemm_xwt_wmma<<<32, 256, 0, stream>>>(sent, w2, nullptr, smask, tmp2,
                                        B_DIM * TS_DIM, D_DIM, D_DIM);
  // tanh/dot: 16 b x 32 v-blocks = 512 blocks
  tanh_dot_kernel<<<B_DIM * (TV_DIM / VPB), 256, 0, stream>>>(
      tmp1, tmp2, wt, vmask, smask, out);
}